// OmegaFlow_40407052320830
// MI455X (gfx1250) — compile-verified
//
#include <hip/hip_runtime.h>

// CDNA5 / gfx1250 Ricci-flow for 512x512 symmetric fp32 graph.
// Core matmul (common = g @ g^T) via V_WMMA_F32_16X16X4_F32, wave32,
// one wave per 16x16 upper-triangular output tile with mirrored writes.

typedef __attribute__((ext_vector_type(2))) float v2f;
typedef __attribute__((ext_vector_type(8))) float v8f;

#define N 512
#define NT 32            // 512/16 tiles per dim
#define TASKS 528        // NT*(NT+1)/2 upper-triangle tiles
#define DT 0.01f
#define EPS 1e-6f
#define TOL2 1e-12f      // (1e-6)^2

// ---- init: copy input graph into ws buffer A, clear stop flag ------------
__global__ void ricci_init(const float* __restrict__ gin, float* __restrict__ A,
                           int* flag) {
    int idx = blockIdx.x * 256 + threadIdx.x;   // grid = 1024 blocks
    if (idx == 0) *flag = 0;
    A[idx] = gin[idx];
}

// ---- deg[i] = sum_j g[i][j] ; one wave per row --------------------------
__global__ void ricci_deg(const float* __restrict__ g, float* __restrict__ deg,
                          const int* __restrict__ flag) {
    if (*flag) return;
    int lane = threadIdx.x & 31;
    int row  = blockIdx.x * 8 + (threadIdx.x >> 5);   // 64 blocks x 8 waves
    float s = 0.f;
    #pragma unroll
    for (int i = 0; i < 16; ++i) s += g[row * N + lane + i * 32];
    #pragma unroll
    for (int off = 16; off; off >>= 1) s += __shfl_xor(s, off, 32);
    if (lane == 0) deg[row] = s;
}

// ---- fused step: WMMA matmul tile + curvature update + symmetrize -------
// One 32-thread wave per upper-triangular 16x16 tile (I<=J); writes the
// mirror tile too. Emits per-wave squared-diff partial for the norm check.
__global__ void ricci_step(const float* __restrict__ g, float* __restrict__ gn,
                           const float* __restrict__ deg,
                           float* __restrict__ partial,
                           const int* __restrict__ flag) {
    // decode triangular task id -> (I, J), I <= J
    int t = blockIdx.x;
    int I = 0, rem = t;
    while (rem >= NT - I) { rem -= NT - I; ++I; }
    int J = I + rem;

    int lane = threadIdx.x;           // blockDim == 32 (full wave, EXEC all 1s)
    int half = lane >> 4;             // 0: lanes 0-15, 1: lanes 16-31
    int lm   = lane & 15;
    int I0 = I * 16, J0 = J * 16;
    int col = J0 + lm;

    if (*flag) {                      // converged/NaN: deterministic copy-through
        #pragma unroll
        for (int r = 0; r < 8; ++r) {
            int row = I0 + r + 8 * half;
            float v = g[row * N + col];
            gn[row * N + col] = v;
            if (I != J) gn[col * N + row] = v;
        }
        return;
    }

    // common[I0+m][J0+n] = sum_k g[I0+m][k] * g[J0+n][k]
    // A 16x4 f32 layout: lanes0-15 -> {K0,K1}, lanes16-31 -> {K2,K3}
    // B 4x16 f32 layout: VGPR r -> K=r (lanes0-15) / K=r+2 (lanes16-31)
    // => both operands are a float2 load at column k0 + 2*half.
    v8f acc = {0.f, 0.f, 0.f, 0.f, 0.f, 0.f, 0.f, 0.f};
    const float* arow = g + (I0 + lm) * N + 2 * half;
    const float* brow = g + (J0 + lm) * N + 2 * half;
    #pragma unroll 4
    for (int kk = 0; kk < N / 4; ++kk) {
        v2f a = *(const v2f*)(arow + kk * 4);
        v2f b = *(const v2f*)(brow + kk * 4);
        acc = __builtin_amdgcn_wmma_f32_16x16x4_f32(
            false, a, false, b, (short)0, acc, false, false);
    }

    // epilogue: curv/ricci/grad, clamp, fused symmetrization, norm partial
    float dj   = deg[col];
    float wfac = (I != J) ? 2.f : 1.f;    // mirrored elements counted twice
    float sq   = 0.f;
    #pragma unroll
    for (int r = 0; r < 8; ++r) {
        int row   = I0 + r + 8 * half;    // C/D layout: VGPR r -> M=r / M=r+8
        float gij = g[row * N + col];
        float di  = deg[row];
        float com = acc[r];
        float curv  = com / (di + dj - com + EPS) - 1.f;
        float ricci = (row != col && gij > 0.f) ? curv : 0.f;
        // grad(tr L^2)[i,j] = 2*deg_i + 2*g_ij off-diag (diag of g is 0)
        float base = gij + DT * (-2.f * ricci + 2.f * gij);
        float ui = base + 2.f * DT * di;  // pre-symmetrize value at (i,j)
        float uj = base + 2.f * DT * dj;  // pre-symmetrize value at (j,i)
        float nv = 0.5f * (fmaxf(ui, 0.f) + fmaxf(uj, 0.f));
        if (row == col) nv = fmaxf(gij, 0.f);   // diagonal: update is exactly 0
        gn[row * N + col] = nv;
        if (I != J) gn[col * N + row] = nv;
        float d = nv - gij;
        sq += wfac * d * d;
    }
    #pragma unroll
    for (int off = 16; off; off >>= 1) sq += __shfl_xor(sq, off, 32);
    if (lane == 0) partial[t] = sq;
}

// ---- convergence check: stop iff !(||diff||^2 >= tol^2) (NaN => stop) ----
__global__ void ricci_check(const float* __restrict__ partial, int* flag) {
    if (*flag) return;
    float s = 0.f;
    for (int i = 0; i < TASKS; ++i) s += partial[i];   // fixed order: deterministic
    if (!(s >= TOL2)) *flag = 1;
}

extern "C" void kernel_launch(void* const* d_in, const int* in_sizes, int n_in,
                              void* d_out, int out_size, void* d_ws, size_t ws_size,
                              hipStream_t stream) {
    const float* gin = (const float*)d_in[0];
    float* out = (float*)d_out;

    float* A   = (float*)d_ws;          // ping buffer, N*N
    float* B   = A + N * N;             // pong buffer, N*N
    float* dg  = B + N * N;             // deg, N
    float* par = dg + N;                // norm partials, TASKS
    int*  flag = (int*)(par + TASKS);   // stop flag

    ricci_init<<<(N * N) / 256, 256, 0, stream>>>(gin, A, flag);

    for (int s = 0; s < 100; ++s) {
        float* cur = (s & 1) ? B : A;
        float* nxt = (s == 99) ? out : ((s & 1) ? A : B);
        ricci_deg <<<N / 8, 256, 0, stream>>>(cur, dg, flag);
        ricci_step<<<TASKS, 32, 0, stream>>>(cur, nxt, dg, par, flag);
        // reference cond: steps 0,1 unconditional; check norm after steps 1..98
        if (s >= 1 && s <= 98)
            ricci_check<<<1, 1, 0, stream>>>(par, flag);
    }
}